// CouchesintermediairesGNN_5497558139182
// MI455X (gfx1250) — compile-verified
//
#include <hip/hip_runtime.h>
#include <hip/hip_bf16.h>

typedef __attribute__((ext_vector_type(2))) float v2f;
typedef __attribute__((ext_vector_type(8))) float v8f;

#define THRESH 10.0f

// ---------------------------------------------------------------------------
// Pass 1: per-edge comb = [one_hot(bucket) | relu(MLP(d))], atomic segment-sum
// into sum_w[src]. One edge per lane; edge streams are coalesced from HBM,
// sum_w (8MB) is L2-resident so the scattered f32 atomics stay on-chip.
// ---------------------------------------------------------------------------
__global__ void edge_pass1(const int* __restrict__ ei, const float* __restrict__ attr,
                           const float* __restrict__ W1, const float* __restrict__ b1,
                           const float* __restrict__ W2, const float* __restrict__ b2,
                           float* __restrict__ sum_w, int E)
{
    int e = blockIdx.x * blockDim.x + threadIdx.x;
    if (e >= E) return;
    int s = ei[e];
    float d = attr[e];

    float acc[10];
#pragma unroll
    for (int j = 0; j < 10; ++j) acc[j] = b2[j];
#pragma unroll 4
    for (int k = 0; k < 64; ++k) {
        float h = fmaxf(fmaf(d, W1[k], b1[k]), 0.0f);
#pragma unroll
        for (int j = 0; j < 10; ++j) acc[j] = fmaf(h, W2[k * 10 + j], acc[j]);
    }
    int bucket = min((int)(d * (10.0f / THRESH)), 9);

    float* swr = sum_w + (size_t)s * 20;
    atomicAdd(&swr[bucket], 1.0f);
#pragma unroll
    for (int j = 0; j < 10; ++j) atomicAdd(&swr[10 + j], fmaxf(acc[j], 0.0f));
}

// ---------------------------------------------------------------------------
// Pass 2: recompute comb (cheaper than a 256MB materialized round-trip),
// normalize by sum_w[src], weight by |a*x_src - (1-a)*x_dst|^b, and
// atomic-accumulate sum_feat directly into out[:,1,:] (it IS the second
// output slot). x (16MB), sum_w (8MB), out (16MB) all fit in the 192MB L2.
// ---------------------------------------------------------------------------
__global__ void edge_pass2(const int* __restrict__ ei, const float* __restrict__ attr,
                           const float* __restrict__ x,
                           const float* __restrict__ ap, const float* __restrict__ bp,
                           const float* __restrict__ W1, const float* __restrict__ b1,
                           const float* __restrict__ W2, const float* __restrict__ b2,
                           const float* __restrict__ sum_w, float* __restrict__ out, int E)
{
    int e = blockIdx.x * blockDim.x + threadIdx.x;
    if (e >= E) return;
    int s = ei[e];
    int t = ei[E + e];
    float d = attr[e];

    float acc[10];
#pragma unroll
    for (int j = 0; j < 10; ++j) acc[j] = b2[j];
#pragma unroll 4
    for (int k = 0; k < 64; ++k) {
        float h = fmaxf(fmaf(d, W1[k], b1[k]), 0.0f);
#pragma unroll
        for (int j = 0; j < 10; ++j) acc[j] = fmaf(h, W2[k * 10 + j], acc[j]);
    }
    int bucket = min((int)(d * (10.0f / THRESH)), 9);

    float av = ap[0], bv = bp[0];
    float cb = 1.0f - av;
    const float* xs  = x + (size_t)s * 40;      // x[s, 0, :]
    const float* xd  = x + (size_t)t * 40;      // x[t, 0, :]
    const float* swr = sum_w + (size_t)s * 20;
    float* oacc = out + (size_t)s * 40 + 20;    // out[s, 1, :]

#pragma unroll
    for (int j = 0; j < 20; ++j) {
        float comb = (j < 10) ? ((j == bucket) ? 1.0f : 0.0f) : fmaxf(acc[j - 10], 0.0f);
        float swj  = swr[j];
        float wt   = (swj != 0.0f) ? (comb / swj) : 0.01f;
        float rho  = powf(fabsf(av * xs[j] - cb * xd[j]), bv);
        atomicAdd(&oacc[j], rho * wt);
    }
}

// ---------------------------------------------------------------------------
// Node update as one WMMA GEMM:
//   D(N x 20) = [X | sum_feat](N x 40) @ Gc(40 x 20)
//   Gc[k][j] = k<20 ? g1[j][k] : g2[j][k-20]
// V_WMMA_F32_16X16X4_F32, K = 40 -> 10 k-steps, two 16-wide column tiles
// (cols 16..19 valid in tile 1). One wave (32 threads) per 16-node tile.
// A layout: lanes 0-15 hold K=k0,k0+1; lanes 16-31 hold K=k0+2,k0+3.
// B mirrors A with N across lanes; C/D: N = lane&15, M = r + 8*(lane>=16).
// The K=20 boundary is 4-aligned, so k-steps 0..4 touch only x/g1 and
// k-steps 5..9 only sum_feat/g2 -> base-pointer choice is compile-time
// (no per-element v_cndmask in front of the WMMAs).
// sum_feat is read back from out[:,1,:]: index k in [20,40) lands at
// out[row*40 + k] exactly.
// ---------------------------------------------------------------------------
__global__ void node_update_wmma(const float* __restrict__ x,
                                 const float* __restrict__ g1, const float* __restrict__ g2,
                                 const float* __restrict__ bias,
                                 float* __restrict__ out, int N)
{
    int lane = threadIdx.x;      // 0..31, one wave per block
    int tile = blockIdx.x;
    int col  = lane & 15;
    int half = lane >> 4;
    int koff = half * 2;         // lanes<16 -> k0,k0+1 ; lanes>=16 -> k0+2,k0+3

    int j0  = col;               // column tile 0: j = 0..15
    int j1  = 16 + col;          // column tile 1: j = 16..31, valid j<20
    int j1c = (j1 < 20) ? j1 : 0;
    bool j1ok = (j1 < 20);

    // B fragments (tile-invariant, kept in registers for the whole wave)
    v2f Bf0[10], Bf1[10];
#pragma unroll
    for (int s = 0; s < 5; ++s) {            // k in [0,20): gamma1
#pragma unroll
        for (int u = 0; u < 2; ++u) {
            int k = 4 * s + koff + u;
            Bf0[s][u] = g1[j0 * 20 + k];
            Bf1[s][u] = j1ok ? g1[j1c * 20 + k] : 0.0f;
        }
    }
#pragma unroll
    for (int s = 5; s < 10; ++s) {           // k in [20,40): gamma2
#pragma unroll
        for (int u = 0; u < 2; ++u) {
            int k2 = 4 * s + koff + u - 20;
            Bf0[s][u] = g2[j0 * 20 + k2];
            Bf1[s][u] = j1ok ? g2[j1c * 20 + k2] : 0.0f;
        }
    }

    int row = tile * 16 + col;
    if (row >= N) row = N - 1;
    const float* xrow = x   + (size_t)row * 40;  // k < 20  -> x[row,0,k]
    const float* srow = out + (size_t)row * 40;  // k >= 20 -> sum_feat at out[row*40+k]

    v8f c0 = {}, c1 = {};
#pragma unroll
    for (int s = 0; s < 5; ++s) {            // A from x
        v2f A;
        A[0] = xrow[4 * s + koff + 0];
        A[1] = xrow[4 * s + koff + 1];
        c0 = __builtin_amdgcn_wmma_f32_16x16x4_f32(false, A, false, Bf0[s],
                                                   (short)0, c0, false, false);
        c1 = __builtin_amdgcn_wmma_f32_16x16x4_f32(false, A, false, Bf1[s],
                                                   (short)0, c1, false, false);
    }
#pragma unroll
    for (int s = 5; s < 10; ++s) {           // A from sum_feat (out[:,1,:])
        v2f A;
        A[0] = srow[4 * s + koff + 0];
        A[1] = srow[4 * s + koff + 1];
        c0 = __builtin_amdgcn_wmma_f32_16x16x4_f32(false, A, false, Bf0[s],
                                                   (short)0, c0, false, false);
        c1 = __builtin_amdgcn_wmma_f32_16x16x4_f32(false, A, false, Bf1[s],
                                                   (short)0, c1, false, false);
    }

    float bias0 = bias[col];
    float bias1 = j1ok ? bias[j1c] : 0.0f;
#pragma unroll
    for (int r = 0; r < 8; ++r) {
        int m    = r + half * 8;
        int node = tile * 16 + m;
        if (node < N) {
            out[(size_t)node * 40 + col] = fmaxf(c0[r] + bias0, 0.0f);
            if (j1ok)
                out[(size_t)node * 40 + j1] = fmaxf(c1[r] + bias1, 0.0f);
        }
    }
}

// ---------------------------------------------------------------------------
extern "C" void kernel_launch(void* const* d_in, const int* in_sizes, int n_in,
                              void* d_out, int out_size, void* d_ws, size_t ws_size,
                              hipStream_t stream)
{
    const float* x    = (const float*)d_in[0];   // [N,2,20]
    const int*   ei   = (const int*)  d_in[1];   // [2,E]
    const float* attr = (const float*)d_in[2];   // [E,1]
    const float* a    = (const float*)d_in[3];
    const float* b    = (const float*)d_in[4];
    const float* g1   = (const float*)d_in[5];   // [20,20]
    const float* g2   = (const float*)d_in[6];   // [20,20]
    const float* bias = (const float*)d_in[7];   // [20]
    const float* W1   = (const float*)d_in[8];   // [64]
    const float* b1   = (const float*)d_in[9];   // [64]
    const float* W2   = (const float*)d_in[10];  // [64,10]
    const float* b2   = (const float*)d_in[11];  // [10]
    float* out = (float*)d_out;                  // [N,2,20]

    int E = in_sizes[2];
    int N = in_sizes[0] / 40;
    float* sum_w = (float*)d_ws;                 // [N,20]

    hipMemsetAsync(out,   0, (size_t)out_size * sizeof(float), stream);
    hipMemsetAsync(sum_w, 0, (size_t)N * 20 * sizeof(float),  stream);

    const int tpb = 256;
    int eblocks = (E + tpb - 1) / tpb;
    edge_pass1<<<eblocks, tpb, 0, stream>>>(ei, attr, W1, b1, W2, b2, sum_w, E);
    edge_pass2<<<eblocks, tpb, 0, stream>>>(ei, attr, x, a, b, W1, b1, W2, b2,
                                            sum_w, out, E);

    int tiles = (N + 15) / 16;
    node_update_wmma<<<tiles, 32, 0, stream>>>(x, g1, g2, bias, out, N);
}